// SelfAttention_1194000908789
// MI455X (gfx1250) — compile-verified
//
#include <hip/hip_runtime.h>

typedef float v2f __attribute__((ext_vector_type(2)));
typedef float v8f __attribute__((ext_vector_type(8)));

#define BATCH 16
#define NTOK  2304      // H*W = 48*48
#define CA    64        // q/k channels
#define CV    256       // v channels
#define NKT   (NTOK/16) // 144 key tiles / query tiles
#define CT_PER_WAVE 8   // column tiles (of 16 channels) per wave; 2 waves cover CV=256

// Flash-attention, fp32 WMMA (16x16x4), wave32.
// One query-tile (16 rows) per workgroup; 2 waves split the 256 V-channels.
__global__ __launch_bounds__(64)
void sagan_attn_f32_wmma(const float* __restrict__ img,
                         const float* __restrict__ f,   // keys   [B,N,CA]
                         const float* __restrict__ g,   // query  [B,N,CA]
                         const float* __restrict__ h,   // values [B,N,CV]
                         const float* __restrict__ gamma,
                         float* __restrict__ out) {
  __shared__ float pl[2][16][17];   // per-wave P bounce buffer (padded)

  const int wg = blockIdx.x;
  const int b  = wg / NKT;
  const int qt = wg % NKT;
  const int q0 = qt * 16;

  const int wid  = threadIdx.x >> 5;   // wave id: 0 or 1
  const int lane = threadIdx.x & 31;
  const int half = lane >> 4;          // 0 or 1
  const int lm   = lane & 15;
  const int c0   = wid * (CT_PER_WAVE * 16);   // this wave's channel base

  // ---- preload G query block as fp32 A-tiles: a[kb] covers K in [4kb, 4kb+4)
  // A layout (16x4 f32): lane l holds row M=l%16, K = 2*(l/16) + {0,1}  -> float2 load
  v2f a[16];
  {
    const float* gq = g + (size_t)(b * NTOK + q0 + lm) * CA + 2 * half;
#pragma unroll
    for (int kb = 0; kb < 16; ++kb)
      a[kb] = *(const v2f*)(gq + 4 * kb);
  }

  // ---- O accumulators: 8 column tiles of 16 channels for this wave
  v8f o[CT_PER_WAVE];
#pragma unroll
  for (int ct = 0; ct < CT_PER_WAVE; ++ct) o[ct] = (v8f){};

  // online-softmax state for the 8 rows this (vgpr,half) slot owns
  float mrow[8], lrow[8];
#pragma unroll
  for (int r = 0; r < 8; ++r) { mrow[r] = -3.0e38f; lrow[r] = 0.f; }

  const float* fb = f + (size_t)b * NTOK * CA;
  const float* hb = h + (size_t)b * NTOK * CV + c0;

  for (int kt = 0; kt < NKT; ++kt) {
    const int k0 = kt * 16;

    // prefetch next key tile (gfx1250 global_prefetch_b8)
    if (kt + 1 < NKT) {
      __builtin_prefetch(fb + (size_t)(k0 + 16 + lm) * CA, 0, 1);
      __builtin_prefetch(hb + (size_t)(k0 + 16 + lm) * CV, 0, 1);
    }

    // ---- S = G * F^T   (16x16 tile, K = 64) : preload all 16 B-tiles, then 16 WMMAs
    v2f bv[16];
    {
      const float* fk = fb + (size_t)(k0 + lm) * CA + 2 * half;
#pragma unroll
      for (int kb = 0; kb < 16; ++kb)
        bv[kb] = *(const v2f*)(fk + 4 * kb);
    }
    v8f s = (v8f){};
#pragma unroll
    for (int kb = 0; kb < 16; ++kb)
      s = __builtin_amdgcn_wmma_f32_16x16x4_f32(false, a[kb], false, bv[kb],
                                                (short)0, s, false, false);

    // ---- online softmax over this key tile; P replaces s
#pragma unroll
    for (int r = 0; r < 8; ++r) {
      float x = s[r];
      float mx = x;
#pragma unroll
      for (int msk = 1; msk <= 8; msk <<= 1)
        mx = fmaxf(mx, __shfl_xor(mx, msk, 32));
      const float newm  = fmaxf(mrow[r], mx);
      const float alpha = __expf(mrow[r] - newm);       // 0 on first tile
      const float p     = __expf(x - newm);
      float ps = p;
#pragma unroll
      for (int msk = 1; msk <= 8; msk <<= 1)
        ps += __shfl_xor(ps, msk, 32);
      lrow[r] = lrow[r] * alpha + ps;
      mrow[r] = newm;
      s[r] = p;
#pragma unroll
      for (int ct = 0; ct < CT_PER_WAVE; ++ct) o[ct][r] *= alpha;
    }

    // ---- P: C/D layout -> LDS (row-major) -> A layout (per-wave buffer)
#pragma unroll
    for (int r = 0; r < 8; ++r)
      pl[wid][r + 8 * half][lm] = s[r];
    // same-wave LDS RAW: wait for DS stores before re-reading (CDNA5 split counter)
    asm volatile("s_wait_dscnt 0x0" ::: "memory");

    v2f pa[4];
#pragma unroll
    for (int kb = 0; kb < 4; ++kb) {
      pa[kb].x = pl[wid][lm][4 * kb + 2 * half];
      pa[kb].y = pl[wid][lm][4 * kb + 2 * half + 1];
    }

    // ---- O += P * H_tile  (K = 16) : per column tile, preload 4 B-tiles then 4 WMMAs
#pragma unroll
    for (int ct = 0; ct < CT_PER_WAVE; ++ct) {
      const float* hc = hb + (size_t)k0 * CV + ct * 16 + lm;
      v2f hv[4];
#pragma unroll
      for (int kb = 0; kb < 4; ++kb) {
        const int krow = 4 * kb + 2 * half;
        hv[kb].x = hc[(size_t)krow * CV];
        hv[kb].y = hc[(size_t)(krow + 1) * CV];
      }
#pragma unroll
      for (int kb = 0; kb < 4; ++kb)
        o[ct] = __builtin_amdgcn_wmma_f32_16x16x4_f32(false, pa[kb], false, hv[kb],
                                                      (short)0, o[ct], false, false);
    }
  }

  // ---- epilogue: out = gamma * (O / l) + img
  const float gam = gamma[0];
#pragma unroll
  for (int ct = 0; ct < CT_PER_WAVE; ++ct) {
#pragma unroll
    for (int r = 0; r < 8; ++r) {
      const int row = q0 + r + 8 * half;
      const size_t idx = (size_t)(b * NTOK + row) * CV + c0 + ct * 16 + lm;
      out[idx] = gam * (o[ct][r] / lrow[r]) + img[idx];
    }
  }
}

extern "C" void kernel_launch(void* const* d_in, const int* in_sizes, int n_in,
                              void* d_out, int out_size, void* d_ws, size_t ws_size,
                              hipStream_t stream) {
  const float* img   = (const float*)d_in[0];
  const float* f     = (const float*)d_in[1];
  const float* g     = (const float*)d_in[2];
  const float* h     = (const float*)d_in[3];
  const float* gamma = (const float*)d_in[4];
  float* out = (float*)d_out;

  dim3 grid(BATCH * NKT);   // 2304 workgroups
  dim3 block(64);           // 2 waves: each owns 128 of the 256 V-channels
  hipLaunchKernelGGL(sagan_attn_f32_wmma, grid, block, 0, stream,
                     img, f, g, h, gamma, out);
}